// RankingLossL1_53326313947168
// MI455X (gfx1250) — compile-verified
//
#include <hip/hip_runtime.h>
#include <cstdint>
#include <cfloat>

#define N_ROWS 50000
#define DIMS   128
#define T_ANCH 1024
#define KSEL   10
#define GAMMA_ 1.0f

#define BA      32                           // anchors per workgroup
#define NRT     128                          // G rows per LDS tile
#define SROW    8                            // row-block splits of N
#define NBLK    ((N_ROWS + SROW - 1) / SROW) // 6250 rows per row-block
#define NTILES  ((NBLK + NRT - 1) / NRT)     // 49 tiles
#define LSTR    132                          // padded LDS row stride (dwords, 16B aligned)
#define THREADS 256
#define RPT     4                            // rows per thread
#define APT     4                            // anchors per thread

// ---------------------------------------------------------------------------
// CDNA5 async copy: 16B global -> LDS via the async engine (ASYNCcnt).
// ---------------------------------------------------------------------------
__device__ __forceinline__ void async_copy16(const float* gsrc, float* ldst) {
  uint32_t l = (uint32_t)(uintptr_t)ldst;   // low 32 bits of generic LDS ptr = LDS byte addr
  asm volatile("global_load_async_to_lds_b128 %0, %1, off"
               :: "v"(l), "v"(gsrc) : "memory");
}

template <int N>
__device__ __forceinline__ void wait_async_le() {
#if __has_builtin(__builtin_amdgcn_s_wait_asynccnt)
  __builtin_amdgcn_s_wait_asynccnt(N);
#else
  asm volatile("s_wait_asynccnt %0" :: "n"(N) : "memory");
#endif
}

// Sorted ascending top-K (tk[K-1] is current worst). One bubble pass re-sorts.
__device__ __forceinline__ void topk_insert(float (&tk)[KSEL], float v) {
  if (v < tk[KSEL - 1]) {
    tk[KSEL - 1] = v;
#pragma unroll
    for (int j = KSEL - 1; j > 0; --j) {
      float lo = fminf(tk[j - 1], tk[j]);
      float hi = fmaxf(tk[j - 1], tk[j]);
      tk[j - 1] = lo;
      tk[j] = hi;
    }
  }
}

// ---------------------------------------------------------------------------
// Kernel 1: gather anchors + positive-pair margin Dm[t] = L1(A1g,A2g) + gamma
// ---------------------------------------------------------------------------
extern "C" __global__ __launch_bounds__(128)
void rl_gather_kernel(const float* __restrict__ out1, const float* __restrict__ out2,
                      const int* __restrict__ a1, const int* __restrict__ a2,
                      float* __restrict__ A1g, float* __restrict__ A2g,
                      float* __restrict__ Dm) {
  __shared__ float red[4];
  const int t = blockIdx.x;
  const int d = threadIdx.x;
  const int i1 = a1[t];
  const int i2 = a2[t];
  float o1 = out1[(size_t)i1 * DIMS + d];
  float o2 = out2[(size_t)i2 * DIMS + d];
  A1g[t * DIMS + d] = o1;
  A2g[t * DIMS + d] = o2;
  float s = fabsf(o1 - o2);
#pragma unroll
  for (int off = 16; off >= 1; off >>= 1) s += __shfl_down(s, off, 32);
  if ((d & 31) == 0) red[d >> 5] = s;
  __syncthreads();
  if (d == 0) Dm[t] = red[0] + red[1] + red[2] + red[3] + GAMMA_;
}

// ---------------------------------------------------------------------------
// Kernel 2: negative mining. One WG = 32 anchors x one row-block x one side.
// Streams 128-row tiles of G through LDS with async double buffering; each
// thread computes a 4 rows x 4 anchors register tile of L1 distances.
// ---------------------------------------------------------------------------
__device__ __forceinline__ void prefetch_tile(const float* __restrict__ G, int rowStart,
                                              int rbEnd, float* buf, int row0, int seg) {
  // 128 rows * 512B = 4096 x 16B chunks; 16 per thread, row step 8 per chunk
#pragma unroll
  for (int i = 0; i < 16; ++i) {
    int row = row0 + 8 * i;
    int grow = rowStart + row;
    grow = (grow < rbEnd) ? grow : (rbEnd - 1);
    async_copy16(G + (size_t)grow * DIMS + seg * 4, buf + row * LSTR + seg * 4);
  }
}

__device__ __forceinline__ void merge_and_store(float (&tk)[KSEL], int lane,
                                                float* __restrict__ dst) {
  float res[KSEL];
#pragma unroll
  for (int k = 0; k < KSEL; ++k) {
    float m = tk[0];
#pragma unroll
    for (int off = 16; off >= 1; off >>= 1) m = fminf(m, __shfl_xor(m, off, 32));
    unsigned long long b = __ballot(tk[0] == m);
    int first = __ffsll(b) - 1;
    if (lane == first) {
#pragma unroll
      for (int j = 0; j < KSEL - 1; ++j) tk[j] = tk[j + 1];
      tk[KSEL - 1] = FLT_MAX;
    }
    res[k] = m;
  }
  if (lane == 0) {
#pragma unroll
    for (int k = 0; k < KSEL; ++k) dst[k] = res[k];
  }
}

extern "C" __global__ __launch_bounds__(THREADS)
void rl_mine_kernel(const float* __restrict__ out1, const float* __restrict__ out2,
                    const float* __restrict__ A1g, const float* __restrict__ A2g,
                    float* __restrict__ topk_out) {
  extern __shared__ float smem[];
  float* gbuf0 = smem;                    // NRT*LSTR
  float* gbuf1 = smem + NRT * LSTR;       // NRT*LSTR
  float* abuf  = smem + 2 * NRT * LSTR;   // BA*LSTR

  const int tid = threadIdx.x;
  const int side = blockIdx.z;            // 0: anchors=A1g vs G=out2 ; 1: A2g vs out1
  const int rb = blockIdx.y;
  const int ab = blockIdx.x;
  const float* G = (side == 0) ? out2 : out1;
  const float* A = (side == 0) ? A1g : A2g;
  const int abStart = ab * BA;
  const int rowStart0 = rb * NBLK;
  const int rbEnd = (rowStart0 + NBLK < N_ROWS) ? (rowStart0 + NBLK) : N_ROWS;

  const int seg  = tid & 31;   // 16B segment within a 512B row
  const int row0 = tid >> 5;   // 0..7

  // async-load the 32 anchor rows (1024 x 16B chunks, 4 per thread), then tile 0
#pragma unroll
  for (int i = 0; i < 4; ++i) {
    int arow = row0 + 8 * i;
    async_copy16(A + (size_t)(abStart + arow) * DIMS + seg * 4,
                 abuf + arow * LSTR + seg * 4);
  }
  prefetch_tile(G, rowStart0, rbEnd, gbuf0, row0, seg);

  const int rg = tid & 31;   // row group (lane):   rows rg + 32*j
  const int ag = tid >> 5;   // wave id:            anchors 4*ag + q
  float tk[APT][KSEL];
#pragma unroll
  for (int q = 0; q < APT; ++q)
#pragma unroll
    for (int k = 0; k < KSEL; ++k) tk[q][k] = FLT_MAX;

  const float* ap = abuf + (APT * ag) * LSTR;

  for (int t = 0; t < NTILES; ++t) {
    float* cur = (t & 1) ? gbuf1 : gbuf0;
    if (t + 1 < NTILES) {
      float* nxt = (t & 1) ? gbuf0 : gbuf1;  // free: last read in previous iter, barrier-protected
      prefetch_tile(G, rowStart0 + (t + 1) * NRT, rbEnd, nxt, row0, seg);
      wait_async_le<16>();                   // in-order: leaves only next tile outstanding
    } else {
      wait_async_le<0>();
    }
    __syncthreads();                         // all waves' portion of tile t visible

    float acc[RPT][APT];
#pragma unroll
    for (int j = 0; j < RPT; ++j)
#pragma unroll
      for (int q = 0; q < APT; ++q) acc[j][q] = 0.f;

    const float* rp = cur + rg * LSTR;
#pragma unroll 4
    for (int d = 0; d < DIMS; d += 4) {
      float4 g[RPT], av[APT];
#pragma unroll
      for (int j = 0; j < RPT; ++j) g[j] = *(const float4*)(rp + (32 * LSTR) * j + d);
#pragma unroll
      for (int q = 0; q < APT; ++q) av[q] = *(const float4*)(ap + LSTR * q + d);
#pragma unroll
      for (int j = 0; j < RPT; ++j)
#pragma unroll
        for (int q = 0; q < APT; ++q) {
          acc[j][q] += fabsf(g[j].x - av[q].x) + fabsf(g[j].y - av[q].y) +
                       fabsf(g[j].z - av[q].z) + fabsf(g[j].w - av[q].w);
        }
    }

    const int tileRow0 = rowStart0 + t * NRT;
#pragma unroll
    for (int j = 0; j < RPT; ++j) {
      if (tileRow0 + rg + 32 * j < rbEnd) {
#pragma unroll
        for (int q = 0; q < APT; ++q) topk_insert(tk[q], acc[j][q]);
      }
    }
    __syncthreads();                         // done reading cur before it is overwritten
  }

  // wave-level merge of 32 sorted lists -> global top-10 per anchor
#pragma unroll
  for (int q = 0; q < APT; ++q) {
    const int tA = abStart + APT * ag + q;
    merge_and_store(tk[q], rg,
                    topk_out + ((size_t)(side * T_ANCH + tA) * SROW + rb) * KSEL);
  }
}

// ---------------------------------------------------------------------------
// Kernel 3: merge row-blocks, loss = sum relu(Dm - d_k) / (T*K). Deterministic.
// ---------------------------------------------------------------------------
extern "C" __global__ __launch_bounds__(256)
void rl_loss_kernel(const float* __restrict__ topk, const float* __restrict__ Dm,
                    float* __restrict__ out) {
  __shared__ float red[256];
  const int tid = threadIdx.x;
  float local = 0.f;
  for (int i = 0; i < (2 * T_ANCH) / 256; ++i) {
    int item = tid + i * 256;             // item = side*T + t
    int t = item & (T_ANCH - 1);
    float dm = Dm[t];
    const float* p = topk + (size_t)item * SROW * KSEL;
    float tk[KSEL];
#pragma unroll
    for (int k = 0; k < KSEL; ++k) tk[k] = FLT_MAX;
    for (int j = 0; j < SROW * KSEL; ++j) topk_insert(tk, p[j]);
#pragma unroll
    for (int k = 0; k < KSEL; ++k) local += fmaxf(dm - tk[k], 0.f);
  }
  red[tid] = local;
  __syncthreads();
  for (int s = 128; s > 0; s >>= 1) {
    if (tid < s) red[tid] += red[tid + s];
    __syncthreads();
  }
  if (tid == 0) out[0] = red[0] / (float)(T_ANCH * KSEL);
}

// ---------------------------------------------------------------------------
extern "C" void kernel_launch(void* const* d_in, const int* in_sizes, int n_in,
                              void* d_out, int out_size, void* d_ws, size_t ws_size,
                              hipStream_t stream) {
  (void)in_sizes; (void)n_in; (void)out_size; (void)ws_size;
  const float* out1 = (const float*)d_in[0];
  const float* out2 = (const float*)d_in[1];
  const int* a1 = (const int*)d_in[2];
  const int* a2 = (const int*)d_in[3];

  float* ws = (float*)d_ws;
  float* A1g  = ws;                          // T*D
  float* A2g  = A1g + T_ANCH * DIMS;         // T*D
  float* Dm   = A2g + T_ANCH * DIMS;         // T
  float* topk = Dm + T_ANCH;                 // 2*T*SROW*KSEL

  rl_gather_kernel<<<T_ANCH, DIMS, 0, stream>>>(out1, out2, a1, a2, A1g, A2g, Dm);

  const size_t lds_bytes = (size_t)(2 * NRT * LSTR + BA * LSTR) * sizeof(float); // ~152KB
  hipFuncSetAttribute(reinterpret_cast<const void*>(rl_mine_kernel),
                      hipFuncAttributeMaxDynamicSharedMemorySize, (int)lds_bytes);
  rl_mine_kernel<<<dim3(T_ANCH / BA, SROW, 2), THREADS, lds_bytes, stream>>>(
      out1, out2, A1g, A2g, topk);

  rl_loss_kernel<<<1, 256, 0, stream>>>(topk, Dm, (float*)d_out);
}